// GA_block_6493990551774
// MI455X (gfx1250) — compile-verified
//
#include <hip/hip_runtime.h>
#include <hip/hip_bf16.h>
#include <math.h>

typedef __attribute__((ext_vector_type(2))) float v2f;
typedef __attribute__((ext_vector_type(8))) float v8f;
typedef __attribute__((ext_vector_type(4))) unsigned int v4u;
typedef __attribute__((ext_vector_type(8))) int v8i;
typedef __attribute__((ext_vector_type(4))) int v4i;

#define Bn    128
#define HWn   1024
#define Cn    384
#define XROWS (HWn + 1)
#define OUTR  257

// TDM tiling for kernel 2: each block stages 128 rows x 32 cols per chunk.
#define TROWS 128              // rows per block (8 waves x 16)
#define CHW   32               // chunk width (floats)
#define NCH   (Cn / CHW)       // 12 chunks
#define PITCH 34               // LDS row pitch: 32 + 2 pad DWORDs (bank-conflict free)

#if defined(__has_builtin)
#if __has_builtin(__builtin_amdgcn_tensor_load_to_lds)
#define USE_TDM 1
#endif
#endif
#ifndef USE_TDM
#define USE_TDM 0
#endif

__device__ __forceinline__ float sigmoidf(float x) { return 1.0f / (1.0f + __expf(-x)); }

// ---------------- Kernel 1: partial column sums of edge_aggregation ----------------
__global__ void k1_ci_acc(const float* __restrict__ ea, float* __restrict__ ci) {
    const int b = blockIdx.x;
    const int slice = blockIdx.y;
    const int c = threadIdx.x;
    const int rows = HWn / 8;
    const float* p = ea + (size_t)b * HWn * Cn + (size_t)slice * rows * Cn + c;
    float s = 0.0f;
    for (int n = 0; n < rows; ++n) {
        if (n + 8 < rows) __builtin_prefetch(p + (size_t)(n + 8) * Cn, 0, 0);
        s += p[(size_t)n * Cn];
    }
    atomicAdd(&ci[b * Cn + c], s);
}

// ---------------- Kernel 1b: ci = sigmoid(mean) ----------------
__global__ void k1b_ci_fin(float* __restrict__ ci) {
    int t = blockIdx.x * blockDim.x + threadIdx.x;
    if (t < Bn * Cn) ci[t] = sigmoidf(ci[t] * (1.0f / (float)HWn));
}

#if USE_TDM
// Issue one TDM descriptor: DMA a TROWS x CHW f32 tile (row stride Cn floats in
// memory) into LDS at lds_off, padding +2 DWORDs after every 32 DWORDs so the
// LDS row pitch is PITCH floats. Tracked by TENSORcnt.
__device__ __forceinline__ void tdm_load_tile(unsigned long long ga, unsigned int lds_off) {
    v4u g0 = { 1u,                                   // count=1 valid descriptor
               lds_off,                              // LDS byte address
               (unsigned int)ga,                     // global_addr[31:0]
               (unsigned int)(ga >> 32) | 0x80000000u };  // addr[56:32] | type=2
    v8i g1 = { (int)((2u << 16)      // data_size = 4B
                   | (1u << 20)      // pad_enable
                   | (4u << 22)      // pad_interval code 4 -> every 32 DWORDs
                   | (1u << 25)),    // pad_amount  code 1 -> 2 DWORDs
               (int)((unsigned)CHW << 16),     // tensor_dim0 low16
               (int)((unsigned)TROWS << 16),   // dim0 hi=0 | tensor_dim1 low16
               (int)((unsigned)CHW << 16),     // dim1 hi=0 | tile_dim0
               (int)TROWS,                     // tile_dim1 | tile_dim2=0
               (int)Cn,                        // tensor_dim0_stride low32
               0, 0 };                         // stride0 hi | tensor_dim1_stride
    v4i z4 = { 0, 0, 0, 0 };
#if __clang_major__ >= 23
    v8i z8 = { 0, 0, 0, 0, 0, 0, 0, 0 };
    __builtin_amdgcn_tensor_load_to_lds(g0, g1, z4, z4, z8, 0);
#else
    __builtin_amdgcn_tensor_load_to_lds(g0, g1, z4, z4, 0);
#endif
}
#endif

// ---------------- Kernel 2: TDM-staged WMMA row-dots + node_aggregation ----------------
// grid Bn*8 blocks of 256 threads (8 waves). Block owns 128 node rows of one
// batch. Wave 0 double-buffers 128x32 f32 tiles into LDS with tensor_load_to_lds
// (fully coalesced DMA, TENSORcnt-tracked); all waves consume their 16-row slice
// with V_WMMA_F32_16X16X4_F32 (ci broadcast across the 16 N columns).
__global__ void k2_ca_na(const float* __restrict__ x, const float* __restrict__ ci,
                         float* __restrict__ ca, float* __restrict__ na) {
    __shared__ float lds_tile[2][TROWS * PITCH];

    const int lane = threadIdx.x & 31;
    const int wv   = threadIdx.x >> 5;
    const int b    = blockIdx.x >> 3;
    const int rowB = (blockIdx.x & 7) * TROWS;   // block's base row
    const int row0 = rowB + (wv << 4);           // this wave's 16-row tile

    const float* nb  = x + (size_t)b * XROWS * Cn + Cn;  // skip cls row
    const float* cib = ci + b * Cn;

    const int m  = lane & 15;          // M row within tile
    const int kk = (lane >> 4) << 1;   // K sub-offset: 0 (lanes 0-15) or 2 (16-31)

    v8f acc = {0.f, 0.f, 0.f, 0.f, 0.f, 0.f, 0.f, 0.f};

#if USE_TDM
    const bool leader = ((__builtin_amdgcn_readfirstlane((int)threadIdx.x) >> 5) == 0);
    if (leader) {
        unsigned long long ga0 =
            (unsigned long long)(uintptr_t)(nb + (size_t)rowB * Cn);
        tdm_load_tile(ga0, (unsigned int)(uintptr_t)&lds_tile[0][0]);
    }
    for (int c = 0; c < NCH; ++c) {
        if (leader) {
            if (c + 1 < NCH) {
                unsigned long long gan = (unsigned long long)(uintptr_t)
                    (nb + (size_t)rowB * Cn + (size_t)(c + 1) * CHW);
                tdm_load_tile(gan, (unsigned int)(uintptr_t)&lds_tile[(c + 1) & 1][0]);
                __builtin_amdgcn_s_wait_tensorcnt(1);  // chunk c retired
            } else {
                __builtin_amdgcn_s_wait_tensorcnt(0);
            }
        }
        __syncthreads();  // publish chunk c to all waves
        const float* lp = &lds_tile[c & 1][(((wv << 4) + m) * PITCH) + kk];
        const float* bp = cib + c * CHW + kk;
#pragma unroll
        for (int k0 = 0; k0 < CHW; k0 += 4) {
            v2f a  = *(const v2f*)(lp + k0);   // ds_load_b64, conflict-free pitch
            v2f bb = *(const v2f*)(bp + k0);
            acc = __builtin_amdgcn_wmma_f32_16x16x4_f32(
                false, a, false, bb, (short)0, acc, false, false);
        }
        __syncthreads();  // chunk c fully consumed before its buffer is reused
    }
#else
    const float* arow = nb + (size_t)(row0 + m) * Cn + kk;
    for (int k0 = 0; k0 < Cn; k0 += 4) {
        v2f a  = *(const v2f*)(arow + k0);
        v2f bb = *(const v2f*)(cib + k0 + kk);
        acc = __builtin_amdgcn_wmma_f32_16x16x4_f32(
            false, a, false, bb, (short)0, acc, false, false);
    }
#endif

    // D layout: VGPR i -> row i (lanes 0-15) / row 8+i (lanes 16-31); all N equal.
    const int sel = lane & 7;
    float rv = acc[0];
    rv = (sel == 1) ? acc[1] : rv;
    rv = (sel == 2) ? acc[2] : rv;
    rv = (sel == 3) ? acc[3] : rv;
    rv = (sel == 4) ? acc[4] : rv;
    rv = (sel == 5) ? acc[5] : rv;
    rv = (sel == 6) ? acc[6] : rv;
    rv = (sel == 7) ? acc[7] : rv;
    // valid in lanes 0-7 (rows 0-7) and 16-23 (rows 8-15)
    float ni = sigmoidf(rv);
    if ((lane & 15) < 8) {
        int row = row0 + ((lane < 16) ? lane : (lane - 8));
        ca[b * HWn + row] = rv;  // pre-sigmoid value, used for argsort
    }

    // broadcast all 16 node_importance values to every lane (wave32 shuffles)
    float niv[16];
#pragma unroll
    for (int mm = 0; mm < 16; ++mm)
        niv[mm] = __shfl(ni, (mm < 8) ? mm : (mm + 8), 32);

    // weighted accumulation: coalesced along c, rows are cache-hot
    for (int c = lane; c < Cn; c += 32) {
        float s = 0.0f;
#pragma unroll
        for (int mm = 0; mm < 16; ++mm)
            s = fmaf(niv[mm], nb[(size_t)(row0 + mm) * Cn + c], s);
        atomicAdd(&na[b * Cn + c], s);
    }
}

// ---------------- Kernel 3: global sum-of-squares of cls and na ----------------
__global__ void k3_norms(const float* __restrict__ cls, const float* __restrict__ na,
                         float* __restrict__ acc) {
    int t = blockIdx.x * blockDim.x + threadIdx.x;  // exactly Bn*Cn threads
    float a = cls[t];
    float c = na[t];
    float s1 = a * a, s2 = c * c;
    for (int off = 16; off > 0; off >>= 1) {
        s1 += __shfl_down(s1, off, 32);
        s2 += __shfl_down(s2, off, 32);
    }
    if ((threadIdx.x & 31) == 0) {
        atomicAdd(&acc[0], s1);
        atomicAdd(&acc[1], s2);
    }
}

// ---------------- Kernel 4: cls_out = cls + (|cls|/|na|) * na ----------------
__global__ void k4_cls(const float* __restrict__ cls, const float* __restrict__ na,
                       const float* __restrict__ acc, float* __restrict__ out) {
    int b = blockIdx.x, c = threadIdx.x;
    float norm = sqrtf(acc[0]) / sqrtf(acc[1]);
    out[(size_t)b * OUTR * Cn + c] = cls[b * Cn + c] + norm * na[b * Cn + c];
}

// ---------------- Kernel 5: bitonic argsort of ca[b,:], gather top quarter ----------------
// Lexicographic (value, index) ascending compare == stable ascending argsort.
__global__ void k5_sort_gather(const float* __restrict__ x, const float* __restrict__ ca,
                               float* __restrict__ out) {
    __shared__ float sv[HWn];
    __shared__ int   si[HWn];
    const int b = blockIdx.x;
    const int tid = threadIdx.x;  // 512 threads

    for (int i = tid; i < HWn; i += 512) {
        sv[i] = ca[b * HWn + i];
        si[i] = i;
    }
    __syncthreads();

    for (int k = 2; k <= HWn; k <<= 1) {
        for (int j = k >> 1; j > 0; j >>= 1) {
            for (int i = tid; i < HWn; i += 512) {
                int l = i ^ j;
                if (l > i) {
                    bool up = ((i & k) == 0);
                    float vi = sv[i], vl = sv[l];
                    int ii = si[i], il = si[l];
                    bool gt = (vi > vl) || (vi == vl && ii > il);
                    if (gt == up) {
                        sv[i] = vl; sv[l] = vi;
                        si[i] = il; si[l] = ii;
                    }
                }
            }
            __syncthreads();
        }
    }

    // gather rows ranked [768, 1024) -> output rows 1..256, coalesced along c
    const float* nb = x + (size_t)b * XROWS * Cn + Cn;
    float* ob = out + (size_t)b * OUTR * Cn + Cn;
    const int base = HWn - HWn / 4;  // 768
    const int total = (HWn / 4) * Cn;
    for (int e = tid; e < total; e += 512) {
        int r = e / Cn;
        int c = e - r * Cn;
        int src = si[base + r];
        ob[(size_t)r * Cn + c] = nb[(size_t)src * Cn + c];
    }
}

extern "C" void kernel_launch(void* const* d_in, const int* in_sizes, int n_in,
                              void* d_out, int out_size, void* d_ws, size_t ws_size,
                              hipStream_t stream) {
    const float* x   = (const float*)d_in[0];  // (B, HW+1, C)
    const float* cls = (const float*)d_in[1];  // (B, 1, C)
    const float* ea  = (const float*)d_in[2];  // (B, HW, C)
    float* out = (float*)d_out;                // (B, 257, C)

    float* ws  = (float*)d_ws;
    float* ci  = ws;                 // Bn*Cn   = 49152 floats
    float* na  = ws + 49152;         // Bn*Cn   = 49152 floats
    float* acc = ws + 98304;         // 2 floats (+pad)
    float* ca  = ws + 98432;         // Bn*HWn  = 131072 floats

    // zero the accumulators (ci, na, acc) every call — graph-capture safe
    hipMemsetAsync(ws, 0, (size_t)98432 * sizeof(float), stream);

    k1_ci_acc<<<dim3(Bn, 8), Cn, 0, stream>>>(ea, ci);
    k1b_ci_fin<<<(Bn * Cn + 255) / 256, 256, 0, stream>>>(ci);
    k2_ca_na<<<Bn * 8, 256, 0, stream>>>(x, ci, ca, na);
    k3_norms<<<(Bn * Cn) / 256, 256, 0, stream>>>(cls, na, acc);
    k4_cls<<<Bn, Cn, 0, stream>>>(cls, na, acc, out);
    k5_sort_gather<<<Bn, 512, 0, stream>>>(x, ca, out);

    (void)in_sizes; (void)n_in; (void)out_size; (void)ws_size;
}